// WaveletScatteringTransform_2035814498800
// MI455X (gfx1250) — compile-verified
//
#include <hip/hip_runtime.h>
#include <math.h>
#include <stdint.h>

// CDNA5 wave32 WMMA fragments for V_WMMA_F32_16X16X4_F32:
//   A (16x4 f32) = 2 VGPRs/lane, B (4x16) = 2 VGPRs/lane, C/D (16x16) = 8 VGPRs/lane
typedef __attribute__((ext_vector_type(2))) float v2f;
typedef __attribute__((ext_vector_type(8))) float v8f;

#define TLEN 16384
#define TD   128

static __device__ __forceinline__ v8f wmma_f32_4(v2f a, v2f b, v8f c) {
  return __builtin_amdgcn_wmma_f32_16x16x4_f32(false, a, false, b, (short)0, c,
                                               false, false);
}

// ---- optional CDNA5 async global->LDS staging (ASYNCcnt-tracked) ----
#if defined(__has_builtin)
#if __has_builtin(__builtin_amdgcn_global_load_async_to_lds_b32) && \
    __has_builtin(__builtin_amdgcn_s_wait_asynccnt)
#define USE_ASYNC_LDS 1
#endif
#endif

#ifdef USE_ASYNC_LDS
typedef __attribute__((address_space(1))) int as1_int;  // global
typedef __attribute__((address_space(3))) int as3_int;  // LDS
static __device__ __forceinline__ void g2lds_b32(const float* g, float* l) {
  // per-lane LDS destination: legal transposing async scatter (ISA 08 §4.4)
  __builtin_amdgcn_global_load_async_to_lds_b32(
      (as1_int*)(uintptr_t)g, (as3_int*)(unsigned)(uintptr_t)l, 0, 0);
}
static __device__ __forceinline__ void g2lds_fence() {
  __builtin_amdgcn_s_wait_asynccnt(0);
}
#else
static __device__ __forceinline__ void g2lds_b32(const float* g, float* l) {
  *l = *g;
}
static __device__ __forceinline__ void g2lds_fence() {}
#endif

// ---------------- table builders ----------------
// F[n][k] = e^{-2pi i nk/128}; Tw[r][c] = e^{-2pi i rc/16384}. Exact integer
// mod reduction keeps angles accurate.
__global__ __launch_bounds__(256) void k_build_tabs(float* __restrict__ Fr,
                                                    float* __restrict__ Fi,
                                                    float* __restrict__ Twr,
                                                    float* __restrict__ Twi) {
  int gid = blockIdx.x * 256 + threadIdx.x;  // 0..16383
  int n = gid >> 7, k = gid & 127;
  int rF = (n * k) & 127;
  float aF = (float)rF * (1.0f / 64.0f);
  Fr[gid] = cospif(aF);
  Fi[gid] = -sinpif(aF);
  int rT = (n * k) & (TLEN - 1);
  float aT = (float)rT * (1.0f / 8192.0f);
  Twr[gid] = cospif(aT);
  Twi[gid] = -sinpif(aT);
}

// g = real(ifft(phi)): g[t] = (1/T) sum_k phi[k] cos(2pi k t / T)
__global__ __launch_bounds__(256) void k_build_g(float* __restrict__ gt,
                                                 const float* __restrict__ phi) {
  int t = blockIdx.x * 256 + threadIdx.x;
  float acc = 0.f;
  for (int k = 0; k < TLEN; ++k) {
    int r = (k * t) & (TLEN - 1);
    acc += phi[k] * cospif((float)r * (1.0f / 8192.0f));
  }
  gt[t] = acc * (1.0f / (float)TLEN);
}

// ---------------- complex GEMM stages (batched 128x128 per signal) ----------
// LEFT:  O = T x D  (T = F, conj(F) if CONJ) ; optional twiddle on D load.
// fwd stage1: <0,0,0>; inv stage2: <1,1,1>.
template <int DCPLX, int CONJ, int USETW>
__global__ __launch_bounds__(256) void k_gemm_left(
    const float* __restrict__ dr, const float* __restrict__ di,
    const float* __restrict__ tbr, const float* __restrict__ tbi,
    const float* __restrict__ twr, const float* __restrict__ twi,
    float* __restrict__ outr, float* __restrict__ outi, float scale) {
  __shared__ float Bsr[16][132];  // [col][k], padded to dodge bank conflicts
  __shared__ float Bsi[16][132];
  const int s   = blockIdx.x >> 3;
  const int ct  = blockIdx.x & 7;
  const int tid = threadIdx.x;
  const int base = s * TLEN;
  const int c0 = ct * 16;
  __builtin_prefetch(tbr, 0, 3);
  if (USETW) {
    for (int e = tid; e < 2048; e += 256) {
      int c = e & 15, k = e >> 4;
      int gi = base + k * TD + c0 + c;
      float vr = dr[gi];
      float vi = DCPLX ? di[gi] : 0.0f;
      int wi_ = k * TD + c0 + c;
      float wr = twr[wi_];
      float wim = CONJ ? -twi[wi_] : twi[wi_];
      Bsr[c][k] = vr * wr - vi * wim;
      Bsi[c][k] = vr * wim + vi * wr;
    }
  } else {
    for (int e = tid; e < 2048; e += 256) {
      int c = e & 15, k = e >> 4;
      int gi = base + k * TD + c0 + c;
      g2lds_b32(&dr[gi], &Bsr[c][k]);
      if (DCPLX) g2lds_b32(&di[gi], &Bsi[c][k]);
    }
    g2lds_fence();
  }
  __syncthreads();
  const int lane = tid & 31;
  const int wave = tid >> 5;
  const int r0 = wave * 16;
  const int lr = lane & 15;
  const int hk = (lane >> 4) << 1;  // lanes 16..31 hold K+2,K+3
  v8f accr = {};
  v8f acci = {};
  for (int k0 = 0; k0 < TD; k0 += 4) {
    int kk = k0 + hk;
    v2f ar  = *(const v2f*)&tbr[(r0 + lr) * TD + kk];
    v2f aiP = *(const v2f*)&tbi[(r0 + lr) * TD + kk];
    v2f aiN = -aiP;                       // single negate, reused
    v2f tiS  = CONJ ? aiN : aiP;          //  sigma*Ti
    v2f tiNS = CONJ ? aiP : aiN;          // -sigma*Ti
    v2f br = *(const v2f*)&Bsr[lr][kk];
    accr = wmma_f32_4(ar, br, accr);      // Tr*Dr
    acci = wmma_f32_4(tiS, br, acci);     // Ti*Dr
    if (DCPLX) {
      v2f bi = *(const v2f*)&Bsi[lr][kk];
      accr = wmma_f32_4(tiNS, bi, accr);  // -Ti*Di
      acci = wmma_f32_4(ar, bi, acci);    //  Tr*Di
    }
  }
  const int hi = lane >> 4;
#pragma unroll
  for (int j = 0; j < 8; ++j) {
    int row = r0 + j + 8 * hi;
    int gi = base + row * TD + c0 + lr;
    outr[gi] = scale * accr[j];
    outi[gi] = scale * acci[j];
  }
}

// RIGHT: O = D x T ; optional twiddle on D load.
// fwd stage2: <1,0,1>; inv stage1: <1,1,0>.
template <int DCPLX, int CONJ, int USETW>
__global__ __launch_bounds__(256) void k_gemm_right(
    const float* __restrict__ dr, const float* __restrict__ di,
    const float* __restrict__ tbr, const float* __restrict__ tbi,
    const float* __restrict__ twr, const float* __restrict__ twi,
    float* __restrict__ outr, float* __restrict__ outi, float scale) {
  __shared__ float Asr[16][132];  // [row][k]
  __shared__ float Asi[16][132];
  const int s   = blockIdx.x >> 3;
  const int rt  = blockIdx.x & 7;
  const int tid = threadIdx.x;
  const int base = s * TLEN;
  const int r0 = rt * 16;
  __builtin_prefetch(tbr, 0, 3);
  if (USETW) {
    for (int e = tid; e < 2048; e += 256) {
      int r = e >> 7, k = e & 127;
      int gi = base + (r0 + r) * TD + k;
      float vr = dr[gi];
      float vi = DCPLX ? di[gi] : 0.0f;
      int wi_ = (r0 + r) * TD + k;
      float wr = twr[wi_];
      float wim = CONJ ? -twi[wi_] : twi[wi_];
      Asr[r][k] = vr * wr - vi * wim;
      Asi[r][k] = vr * wim + vi * wr;
    }
  } else {
    for (int e = tid; e < 2048; e += 256) {
      int r = e >> 7, k = e & 127;
      int gi = base + (r0 + r) * TD + k;
      g2lds_b32(&dr[gi], &Asr[r][k]);
      if (DCPLX) g2lds_b32(&di[gi], &Asi[r][k]);
    }
    g2lds_fence();
  }
  __syncthreads();
  const int lane = tid & 31;
  const int wave = tid >> 5;
  const int c0 = wave * 16;
  const int lr = lane & 15;
  const int hk = (lane >> 4) << 1;
  v8f accr = {};
  v8f acci = {};
  for (int k0 = 0; k0 < TD; k0 += 4) {
    int kk = k0 + hk;
    v2f ar = *(const v2f*)&Asr[lr][kk];
    v2f br, biP;
    br.x  = tbr[kk * TD + c0 + lr];
    br.y  = tbr[(kk + 1) * TD + c0 + lr];
    biP.x = tbi[kk * TD + c0 + lr];
    biP.y = tbi[(kk + 1) * TD + c0 + lr];
    v2f biN = -biP;
    v2f tiS  = CONJ ? biN : biP;          //  sigma*Ti
    v2f tiNS = CONJ ? biP : biN;          // -sigma*Ti
    accr = wmma_f32_4(ar, br, accr);      // Dr*Tr
    acci = wmma_f32_4(ar, tiS, acci);     // Dr*Ti
    if (DCPLX) {
      v2f ai = *(const v2f*)&Asi[lr][kk];
      accr = wmma_f32_4(ai, tiNS, accr);  // -Di*Ti
      acci = wmma_f32_4(ai, br, acci);    //  Di*Tr
    }
  }
  const int hi = lane >> 4;
#pragma unroll
  for (int j = 0; j < 8; ++j) {
    int row = r0 + j + 8 * hi;
    int gi = base + row * TD + c0 + lr;
    outr[gi] = scale * accr[j];
    outi[gi] = scale * acci[j];
  }
}

// ---------------- pointwise / modulus ----------------
// spectrum storage index j <-> true frequency k: k = (j>>7) + ((j&127)<<7)
__global__ __launch_bounds__(256) void k_mul_psi1(
    const float* __restrict__ xr, const float* __restrict__ xi,
    const float* __restrict__ psi1, float* __restrict__ yr,
    float* __restrict__ yi) {
  int gid = blockIdx.x * 256 + threadIdx.x;   // 256 signals * 16384
  int sidx = gid >> 14, j = gid & (TLEN - 1);
  int b = sidx >> 6, n1 = sidx & 63;
  int kt = ((j & 127) << 7) | (j >> 7);
  float f = psi1[n1 * TLEN + kt];
  yr[gid] = xr[b * TLEN + j] * f;
  yi[gid] = xi[b * TLEN + j] * f;
}

__global__ __launch_bounds__(256) void k_mul_psi2(
    const float* __restrict__ ur, const float* __restrict__ ui,
    const float* __restrict__ psi2, int pbase, float* __restrict__ yr,
    float* __restrict__ yi) {
  int gid = blockIdx.x * 256 + threadIdx.x;
  int pl = gid >> 14, j = gid & (TLEN - 1);
  int p = pbase + pl;
  int src = p >> 3, n2 = p & 7;
  int kt = ((j & 127) << 7) | (j >> 7);
  float f = psi2[n2 * TLEN + kt];
  yr[gid] = ur[src * TLEN + j] * f;
  yi[gid] = ui[src * TLEN + j] * f;
}

__global__ __launch_bounds__(256) void k_abs(const float* __restrict__ cr,
                                             const float* __restrict__ ci,
                                             float* __restrict__ u) {
  int gid = blockIdx.x * 256 + threadIdx.x;
  float a = cr[gid], b = ci[gid];
  u[gid] = sqrtf(a * a + b * b);
}

// |c| stored transposed: U2T[t][path] for the S2 GEMM (t is K-dim)
__global__ __launch_bounds__(256) void k_abs_t(const float* __restrict__ cr,
                                               const float* __restrict__ ci,
                                               float* __restrict__ u2t) {
  int gid = blockIdx.x * 256 + threadIdx.x;
  int pl = gid >> 14, t = gid & (TLEN - 1);
  float a = cr[gid], b = ci[gid];
  u2t[t * 256 + pl] = sqrtf(a * a + b * b);
}

// ---------------- lowpass via spectral fold (S0, S1) ----------------
// y[m] = (1/T) sum_k phi[k] * Re(H[k] e^{+2pi i k m/64})
__global__ __launch_bounds__(256) void k_fold(const float* __restrict__ hr,
                                              const float* __restrict__ hi,
                                              const float* __restrict__ phi,
                                              float* __restrict__ out, int mode) {
  __shared__ float red[256];
  const int m = blockIdx.x;
  const int sidx = blockIdx.y;
  const int tid = threadIdx.x;
  const int base = sidx * TLEN;
  float acc = 0.f;
  for (int j = tid; j < TLEN; j += 256) {
    int kt = ((j & 127) << 7) | (j >> 7);
    float p = phi[kt];
    int r = (kt * m) & 63;
    float cc = cospif((float)r * (1.0f / 32.0f));
    float ss = sinpif((float)r * (1.0f / 32.0f));
    acc += p * (hr[base + j] * cc - hi[base + j] * ss);
  }
  red[tid] = acc;
  __syncthreads();
  for (int off = 128; off > 0; off >>= 1) {
    if (tid < off) red[tid] += red[tid + off];
    __syncthreads();
  }
  if (tid == 0) {
    int row = (mode == 0) ? (sidx * 577)
                          : ((sidx >> 6) * 577 + 1 + (sidx & 63));
    out[row * 64 + m] = red[0] * (1.0f / (float)TLEN);
  }
}

// ---------------- S2 lowpass GEMM: S2[m][p] = sum_t g[(256m-t)&16383]*U2T[t][p]
__global__ __launch_bounds__(256) void k_gemm_s2(const float* __restrict__ gt,
                                                 const float* __restrict__ u2t,
                                                 float* __restrict__ out,
                                                 int pbase) {
  const int tid = threadIdx.x;
  const int lane = tid & 31;
  const int wave = tid >> 5;
  const int tileid = blockIdx.x * 8 + wave;  // 64 tiles: 4 row x 16 col
  const int rt = tileid & 3;
  const int ctile = tileid >> 2;
  const int lr = lane & 15;
  const int hk = (lane >> 4) << 1;
  const int mrow = rt * 16 + lr;      // A-fragment row (output m)
  const int col = ctile * 16 + lr;    // B-fragment col (path in chunk)
  v8f acc = {};
  for (int k0 = 0; k0 < TLEN; k0 += 4) {
    int kk = k0 + hk;
    v2f a, b;
    a.x = gt[(256 * mrow - kk) & (TLEN - 1)];
    a.y = gt[(256 * mrow - kk - 1) & (TLEN - 1)];
    b.x = u2t[kk * 256 + col];
    b.y = u2t[(kk + 1) * 256 + col];
    acc = wmma_f32_4(a, b, acc);
  }
  const int hi = lane >> 4;
#pragma unroll
  for (int j = 0; j < 8; ++j) {
    int mm = rt * 16 + j + 8 * hi;
    int p = pbase + col;
    int b_ = p >> 9, n1 = (p >> 3) & 63, n2 = p & 7;
    float v = (8 * n2 > n1) ? acc[j] : 0.0f;  // mask: xi2[n2] < xi1[n1]
    out[(b_ * 577 + 65 + n1 * 8 + n2) * 64 + mm] = v;
  }
}

extern "C" void kernel_launch(void* const* d_in, const int* in_sizes, int n_in,
                              void* d_out, int out_size, void* d_ws,
                              size_t ws_size, hipStream_t stream) {
  (void)in_sizes; (void)n_in; (void)out_size; (void)ws_size;
  const float* x    = (const float*)d_in[0];  // (4,1,16384)
  const float* psi1 = (const float*)d_in[1];  // (64,16384)
  const float* psi2 = (const float*)d_in[2];  // (8,16384)
  const float* phi  = (const float*)d_in[3];  // (16384,)
  float* out = (float*)d_out;                 // (4,1,577,64)
  float* w = (float*)d_ws;
  // ws layout (floats) — total ~33.8M floats (~135 MB)
  float* Fr   = w;
  float* Fi   = w + 16384;
  float* Twr  = w + 32768;
  float* Twi  = w + 49152;
  float* gtab = w + 65536;
  float* Xr   = w + 81920;      // 4*16384
  float* Xi   = w + 147456;
  float* U1   = w + 212992;     // 256*16384
  float* U1hr = w + 4407296;
  float* U1hi = w + 8601600;
  float* Yr   = w + 12795904;
  float* Yi   = w + 16990208;
  float* Ar   = w + 21184512;
  float* Ai   = w + 25378816;
  float* U2T  = w + 29573120;   // 16384 x 256

  const float INV = 1.0f / 16384.0f;

  k_build_tabs<<<64, 256, 0, stream>>>(Fr, Fi, Twr, Twi);
  k_build_g<<<64, 256, 0, stream>>>(gtab, phi);

  // ---- forward FFT of x (4 real signals): A=F*D, then C=(A o Tw)*F ----
  k_gemm_left<0, 0, 0><<<4 * 8, 256, 0, stream>>>(x, x, Fr, Fi, Twr, Twi, Ar, Ai, 1.f);
  k_gemm_right<1, 0, 1><<<4 * 8, 256, 0, stream>>>(Ar, Ai, Fr, Fi, Twr, Twi, Xr, Xi, 1.f);
  // S0
  k_fold<<<dim3(64, 4), 256, 0, stream>>>(Xr, Xi, phi, out, 0);

  // ---- first order: Y1 = X*psi1, u1 = ifft(Y1), U1=|u1|, U1h=fft(U1) ----
  k_mul_psi1<<<16384, 256, 0, stream>>>(Xr, Xi, psi1, Yr, Yi);
  k_gemm_right<1, 1, 0><<<256 * 8, 256, 0, stream>>>(Yr, Yi, Fr, Fi, Twr, Twi, Ar, Ai, 1.f);
  k_gemm_left<1, 1, 1><<<256 * 8, 256, 0, stream>>>(Ar, Ai, Fr, Fi, Twr, Twi, Yr, Yi, INV);
  k_abs<<<16384, 256, 0, stream>>>(Yr, Yi, U1);
  k_gemm_left<0, 0, 0><<<256 * 8, 256, 0, stream>>>(U1, U1, Fr, Fi, Twr, Twi, Ar, Ai, 1.f);
  k_gemm_right<1, 0, 1><<<256 * 8, 256, 0, stream>>>(Ar, Ai, Fr, Fi, Twr, Twi, U1hr, U1hi, 1.f);
  // S1
  k_fold<<<dim3(64, 256), 256, 0, stream>>>(U1hr, U1hi, phi, out, 1);

  // ---- second order, 8 chunks of 256 paths ----
  for (int c = 0; c < 8; ++c) {
    int pbase = c * 256;
    k_mul_psi2<<<16384, 256, 0, stream>>>(U1hr, U1hi, psi2, pbase, Yr, Yi);
    k_gemm_right<1, 1, 0><<<256 * 8, 256, 0, stream>>>(Yr, Yi, Fr, Fi, Twr, Twi, Ar, Ai, 1.f);
    k_gemm_left<1, 1, 1><<<256 * 8, 256, 0, stream>>>(Ar, Ai, Fr, Fi, Twr, Twi, Yr, Yi, INV);
    k_abs_t<<<16384, 256, 0, stream>>>(Yr, Yi, U2T);
    k_gemm_s2<<<8, 256, 0, stream>>>(gtab, U2T, out, pbase);
  }
}